// GreedyInvasionMemory_12326556139955
// MI455X (gfx1250) — compile-verified
//
#include <hip/hip_runtime.h>

// GreedyInvasionMemory scan for MI455X (gfx1250).
// Single persistent workgroup (8 wave32 waves); Sqq / Sqv^T / J live in the
// 320KB WGP LDS (192KB used). All 128x128 matvecs run on the XDL pipe via
// V_WMMA_F32_16X16X4_F32. Trace quantities (tr(J Sqq J^T), tr(J Sqv),
// tr(Svv)) are maintained by exact closed-form scalar recurrences, removing
// the reference's per-step 128^3 matmuls entirely.

#define SEQ_LEN  2048
#define D        128
#define NTHREADS 256   // 8 waves of 32 on CDNA5

typedef float v2f __attribute__((ext_vector_type(2)));
typedef float v8f __attribute__((ext_vector_type(8)));

// y[strip*16 .. strip*16+15] = M(128x128, row-major in LDS) * x(128, LDS)
// One wave per strip. A tile = 16x4 rows of M; B = x chunk broadcast across
// all 16 N-columns (B VGPR0 <-> K = 2*hi, VGPR1 <-> K = 2*hi+1, mirroring A).
// All C columns are identical; lanes with N==0 (lane 0: rows 0-7, lane 16:
// rows 8-15) write the result strip back to LDS.
__device__ __forceinline__ void matvec_strip_wmma(const float* M, const float* x,
                                                  float* y, int strip, int lane) {
    const int row = lane & 15;
    const int hi  = lane >> 4;          // 0: lanes 0-15, 1: lanes 16-31
    const float* mrow = M + (size_t)(strip * 16 + row) * D;
    v8f acc = {};
#pragma unroll 4
    for (int c = 0; c < 32; ++c) {      // K = 128 in chunks of 4
        const int kb = c * 4 + hi * 2;
        v2f a; a.x = mrow[kb]; a.y = mrow[kb + 1];
        v2f b; b.x = x[kb];    b.y = x[kb + 1];
        // (neg_a, A, neg_b, B, c_mod, C, reuse_a, reuse_b)
        acc = __builtin_amdgcn_wmma_f32_16x16x4_f32(false, a, false, b,
                                                    (short)0, acc, false, false);
    }
    if (row == 0) {
#pragma unroll
        for (int r = 0; r < 8; ++r) y[strip * 16 + hi * 8 + r] = acc[r];
    }
}

__global__ void __launch_bounds__(NTHREADS, 1)
gim_scan_kernel(const float* __restrict__ Q, const float* __restrict__ K,
                const float* __restrict__ V, float* __restrict__ out) {
    extern __shared__ float smem[];
    float* sJ    = smem;                 // [D*D]  J      (d_v x d_k)
    float* sSqq  = sJ   + D * D;         // [D*D]  Sqq    (d_k x d_k)
    float* sSvq  = sSqq + D * D;         // [D*D]  Sqv^T  (d_v x d_k)
    float* vecs  = sSvq + D * D;
    float* sQ = vecs;         float* sK = vecs + D;     float* sV = vecs + 2 * D;
    float* sU = vecs + 3 * D; float* sT = vecs + 4 * D; // u = J q ; t = Sqq k
    float* sR = vecs + 5 * D; float* sW = vecs + 6 * D; // r = Sqv^T k ; w = J t
    float* sDots  = vecs + 7 * D;        // vv, uu, uv, kt, rv, vw
    float* sState = sDots + 8;           // A_JJ, s_J, trSvv

    const int tid  = threadIdx.x;
    const int lane = tid & 31;
    const int wave = tid >> 5;

    // deterministic init (harness poisons nothing back between replays)
    for (int e = tid; e < 3 * D * D; e += NTHREADS) sJ[e] = 0.0f;
    if (tid < 3) sState[tid] = 0.0f;
    __syncthreads();

    for (int l = 1; l <= SEQ_LEN; ++l) {
        const float fl    = (float)l;
        const float inv_l = 1.0f / fl;
        const float scale = (fl - 1.0f) * inv_l;
        const int   base  = (l - 1) * D;

        if (tid < D) {
            sQ[tid] = Q[base + tid];
            sK[tid] = K[base + tid];
            sV[tid] = V[base + tid];
            if (l < SEQ_LEN) {          // global_prefetch_b8 of next row
                __builtin_prefetch(&Q[base + D + tid], 0, 1);
                __builtin_prefetch(&K[base + D + tid], 0, 1);
                __builtin_prefetch(&V[base + D + tid], 0, 1);
            }
        }
        __syncthreads();

        // rank-1 running-mean updates: Sqq += qq^T, Sqv^T += vq^T
        for (int e = tid; e < D * D; e += NTHREADS) {
            const int i = e >> 7, j = e & 127;
            sSqq[e] = scale * sSqq[e] + (sQ[i] * sQ[j]) * inv_l;
            sSvq[e] = scale * sSvq[e] + (sV[i] * sQ[j]) * inv_l;
        }
        __syncthreads();

        // WMMA matvec group 1 (independent): u = J q, t = Sqq k, r = Sqv^T k
        matvec_strip_wmma(sJ,   sQ, sU, wave, lane);
        matvec_strip_wmma(sSqq, sK, sT, wave, lane);
        matvec_strip_wmma(sSvq, sK, sR, wave, lane);
        __syncthreads();
        // group 2 (needs t): w = J t
        matvec_strip_wmma(sJ, sT, sW, wave, lane);
        __syncthreads();

        // six dot products over D=128, wave 0 only (4 elems/lane + xor-shuffle)
        if (wave == 0) {
            float vv = 0, uu = 0, uv = 0, kt = 0, rv = 0, vw = 0;
            for (int i = lane; i < D; i += 32) {
                const float vi = sV[i], ki = sK[i], ui = sU[i];
                const float ti = sT[i], ri = sR[i], wi2 = sW[i];
                vv += vi * vi; uu += ui * ui; uv += ui * vi;
                kt += ki * ti; rv += ri * vi; vw += vi * wi2;
            }
            for (int m = 16; m >= 1; m >>= 1) {
                vv += __shfl_xor(vv, m, 32); uu += __shfl_xor(uu, m, 32);
                uv += __shfl_xor(uv, m, 32); kt += __shfl_xor(kt, m, 32);
                rv += __shfl_xor(rv, m, 32); vw += __shfl_xor(vw, m, 32);
            }
            if (lane == 0) {
                sDots[0] = vv; sDots[1] = uu; sDots[2] = uv;
                sDots[3] = kt; sDots[4] = rv; sDots[5] = vw;
            }
        }
        __syncthreads();

        // scalar logic, replicated identically on every thread
        const float vv = sDots[0], uu = sDots[1], uv = sDots[2];
        const float kt = sDots[3], rv = sDots[4], vw = sDots[5];
        const float trSvv = scale * sState[2] + vv * inv_l;
        const float A_JJ  = scale * sState[0] + uu * inv_l;  // tr(J Sqq J^T)
        const float s_J   = scale * sState[1] + uv * inv_l;  // tr(J Sqv)
        const float s_l   = rv;                              // k^T Sqv v
        const float A_ll  = vv * kt;                         // (v.v)(k Sqq k)
        const float A_Jl  = vw;                              // v^T J Sqq k
        const bool  first = (l == 1);
        const float A_JJ_s  = (first || A_JJ == 0.0f) ? 1.0f : A_JJ;
        const float A_ll_s  = (first || A_ll == 0.0f) ? 1.0f : A_ll;
        const float denom   = A_JJ * A_ll - A_Jl * A_Jl;
        const float denom_s = (first || denom == 0.0f) ? 1.0f : denom;
        const float margin  = s_l - A_Jl * (s_J / A_JJ_s);
        const float wf = (A_ll * s_J - A_Jl * s_l) / denom_s;
        const float wi = (A_JJ * s_l - A_Jl * s_J) / denom_s;
        const float wf_c = (wi <= 0.0f) ? (s_J / A_JJ_s)
                                        : ((wf <= 0.0f) ? 0.0f : wf);
        const float wi_c = (wi <= 0.0f) ? 0.0f
                                        : ((wf <= 0.0f) ? (s_l / A_ll_s) : wi);
        const bool  do_upd = margin > 0.0f;
        // J_new = a*J + b*(v k^T); first step forces (0,1), no-update (1,0)
        const float a = first ? 0.0f : (do_upd ? wf_c : 1.0f);
        const float b = first ? 1.0f : (do_upd ? wi_c : 0.0f);
        const float s_Jn  = a * s_J + b * s_l;
        const float A_JJn = a * a * A_JJ + 2.0f * a * b * A_Jl + b * b * A_ll;
        const float cost  = 0.5f * trSvv - s_Jn + 0.5f * A_JJn;
        const bool  updated = first || do_upd;
        __syncthreads();   // everyone has read sState/sDots before overwrite
        if (tid == 0) {
            sState[0] = A_JJn; sState[1] = s_Jn; sState[2] = trSvv;
            out[l - 1]           = cost;
            out[SEQ_LEN + l - 1] = updated ? 1.0f : 0.0f;
        }
        if (updated) {     // uniform branch (all threads agree)
            for (int e = tid; e < D * D; e += NTHREADS) {
                const int i = e >> 7, j = e & 127;
                sJ[e] = a * sJ[e] + b * sV[i] * sK[j];
            }
        }
        __syncthreads();
    }

    // final J (d_v x d_k row-major), third output
    for (int e = tid; e < D * D; e += NTHREADS) out[2 * SEQ_LEN + e] = sJ[e];
}

extern "C" void kernel_launch(void* const* d_in, const int* in_sizes, int n_in,
                              void* d_out, int out_size, void* d_ws, size_t ws_size,
                              hipStream_t stream) {
    (void)in_sizes; (void)n_in; (void)out_size; (void)d_ws; (void)ws_size;
    const float* Q = (const float*)d_in[0];
    const float* K = (const float*)d_in[1];
    const float* V = (const float*)d_in[2];
    float* out = (float*)d_out;

    const size_t shmem = (size_t)(3 * D * D + 8 * D + 16) * sizeof(float); // ~196KB of 320KB WGP LDS
    // >64KB dynamic LDS opt-in; deterministic, not a stream op (capture-safe)
    (void)hipFuncSetAttribute((const void*)gim_scan_kernel,
                              hipFuncAttributeMaxDynamicSharedMemorySize,
                              (int)shmem);
    gim_scan_kernel<<<1, NTHREADS, shmem, stream>>>(Q, K, V, out);
}